// GraphSAGE_27419071218491
// MI455X (gfx1250) — compile-verified
//
#include <hip/hip_runtime.h>

#define N_NODES 16384
#define N_EDGES 524288
#define HID     16
#define R_OUT   237

typedef __attribute__((ext_vector_type(2))) float v2f;
typedef __attribute__((ext_vector_type(8))) float v8f;

// ---------------------------------------------------------------------------
// Pack a [K,16] row-major B matrix into the per-lane V_WMMA_F32_16X16X4_F32
// B-fragment layout: for each K-step (4 rows), 64 floats ordered so lane L's
// two B values (VGPR0/VGPR1) sit at PB[step*64 + L*2 + {0,1}].
//   VGPR0 (j=0): B[k + 2*(L>>4),     L&15]
//   VGPR1 (j=1): B[k + 2*(L>>4) + 1, L&15]
// ---------------------------------------------------------------------------
__global__ void pack_b_kernel(const float* __restrict__ B,
                              float* __restrict__ PB, int K) {
  int i = blockIdx.x * blockDim.x + threadIdx.x;
  int n = (K >> 2) * 64;
  if (i >= n) return;
  int kstep    = i >> 6;        // k / 4
  int lanepair = i & 63;        // lane*2 + j
  int lane     = lanepair >> 1;
  int j        = lanepair & 1;
  int col      = lane & 15;
  int krow     = (kstep << 2) + 2 * (lane >> 4) + j;
  PB[i] = B[(size_t)krow * HID + col];
}

// ---------------------------------------------------------------------------
// Dual GEMM: Cl = A @ Bl, Cr = A @ Br. A is [M,K] row-major; PBl/PBr are the
// packed B streams from pack_b_kernel. One 16-row M-tile per block; waves
// split K and reduce through LDS. Each A fragment (one global_load_b64 per
// lane) feeds BOTH WMMAs so A is read exactly once.
// ---------------------------------------------------------------------------
__global__ void dual_gemm16_wmma(const float* __restrict__ A,
                                 const float* __restrict__ PBl,
                                 const float* __restrict__ PBr,
                                 float* __restrict__ Cl,
                                 float* __restrict__ Cr,
                                 int K) {
  const int m0     = blockIdx.x * 16;
  const int lane   = threadIdx.x & 31;
  const int wave   = threadIdx.x >> 5;
  const int nwaves = blockDim.x >> 5;
  const int row    = lane & 15;   // M index (A) / N column (B, C)
  const int khalf  = lane >> 4;   // 0: K pair {0,1}, 1: K pair {2,3}

  const int kPerWave = K / nwaves;
  const int kBeg     = wave * kPerWave;
  const int steps    = kPerWave >> 2;

  v8f cl = {};
  v8f cr = {};

  const float* aPtr  = A + (size_t)(m0 + row) * (size_t)K + kBeg + 2 * khalf;
  const float* blPtr = PBl + (size_t)(kBeg >> 2) * 64 + lane * 2;
  const float* brPtr = PBr + (size_t)(kBeg >> 2) * 64 + lane * 2;

  #pragma unroll 8
  for (int s = 0; s < steps; ++s) {
    v2f a  = *(const v2f*)aPtr;    // A[row, k..k+1] (lanes 0-15) / k+2..k+3 (16-31)
    v2f bl = *(const v2f*)blPtr;   // packed B fragment, one b64, fully coalesced
    v2f br = *(const v2f*)brPtr;
    cl = __builtin_amdgcn_wmma_f32_16x16x4_f32(false, a, false, bl, (short)0, cl,
                                               false, false);
    cr = __builtin_amdgcn_wmma_f32_16x16x4_f32(false, a, false, br, (short)0, cr,
                                               false, false);
    aPtr  += 4;
    blPtr += 64;
    brPtr += 64;
  }

  // Cross-wave K reduction through LDS (skipped for single-wave launches).
  __shared__ float red[8 * 512];  // 8 waves * (2 matrices * 8 vgprs * 32 lanes)
  if (nwaves > 1) {
    #pragma unroll
    for (int j = 0; j < 8; ++j) {
      red[wave * 512 +       j * 32 + lane] = cl[j];
      red[wave * 512 + 256 + j * 32 + lane] = cr[j];
    }
    __syncthreads();
    if (wave == 0) {
      #pragma unroll
      for (int j = 0; j < 8; ++j) {
        float sl = 0.0f, sr = 0.0f;
        for (int w = 0; w < nwaves; ++w) {
          sl += red[w * 512 +       j * 32 + lane];
          sr += red[w * 512 + 256 + j * 32 + lane];
        }
        cl[j] = sl;
        cr[j] = sr;
      }
    }
  }

  if (wave == 0) {
    // C layout: vgpr j holds (M=j, N=lane) lanes 0-15; (M=j+8, N=lane-16) 16-31.
    #pragma unroll
    for (int j = 0; j < 8; ++j) {
      const int r = (lane < 16) ? j : (j + 8);
      const int c = lane & 15;
      Cl[(size_t)(m0 + r) * HID + c] = cl[j];
      Cr[(size_t)(m0 + r) * HID + c] = cr[j];
    }
  }
}

// ---------------------------------------------------------------------------
__global__ void zero_f32(float* __restrict__ p, int n) {
  int i = blockIdx.x * blockDim.x + threadIdx.x;
  if (i < n) p[i] = 0.0f;
}

__global__ void degree_kernel(const long long* __restrict__ tgt,
                              float* __restrict__ deg, int E) {
  int e = blockIdx.x * blockDim.x + threadIdx.x;
  if (e < E) unsafeAtomicAdd(&deg[tgt[e]], 1.0f);
}

// agg[tgt, f] += h[src, f]  (one thread per (edge, feature))
__global__ void scatter_add_kernel(const float* __restrict__ h,
                                   const long long* __restrict__ src,
                                   const long long* __restrict__ tgt,
                                   float* __restrict__ agg, int E) {
  int i = blockIdx.x * blockDim.x + threadIdx.x;
  if (i >= E * HID) return;
  int e = i >> 4;
  int f = i & (HID - 1);
  long long s = src[e];
  long long t = tgt[e];
  unsafeAtomicAdd(&agg[t * HID + f], h[s * HID + f]);
}

// out = [relu]( agg/max(deg,1) + b + hr )
__global__ void combine_kernel(const float* __restrict__ agg,
                               const float* __restrict__ deg,
                               const float* __restrict__ b,
                               const float* __restrict__ hr,
                               float* __restrict__ out, int do_relu, int n) {
  int i = blockIdx.x * blockDim.x + threadIdx.x;
  if (i >= n) return;
  int node = i >> 4;
  int f = i & (HID - 1);
  float d = deg[node];
  d = d > 1.0f ? d : 1.0f;
  float v = agg[i] / d + b[f] + hr[i];
  out[i] = do_relu ? fmaxf(v, 0.0f) : v;
}

// out[i, r] = bfc[r] + sum_k h[i,k] * Wfc[k,r]   (bandwidth-bound store)
__global__ void fc_kernel(const float* __restrict__ h,
                          const float* __restrict__ W,
                          const float* __restrict__ b,
                          float* __restrict__ out) {
  int i = blockIdx.x * blockDim.x + threadIdx.x;
  if (i >= N_NODES * R_OUT) return;
  int node = i / R_OUT;
  int r = i - node * R_OUT;
  float acc = b[r];
  #pragma unroll
  for (int k = 0; k < HID; ++k) acc += h[node * HID + k] * W[k * R_OUT + r];
  out[i] = acc;
}

// ---------------------------------------------------------------------------
extern "C" void kernel_launch(void* const* d_in, const int* in_sizes, int n_in,
                              void* d_out, int out_size, void* d_ws, size_t ws_size,
                              hipStream_t stream) {
  const float*     x   = (const float*)d_in[0];
  const long long* ei  = (const long long*)d_in[1];  // int64 [2, E]
  const float*     W1l = (const float*)d_in[2];
  const float*     b1  = (const float*)d_in[3];
  const float*     W1r = (const float*)d_in[4];
  const float*     W2l = (const float*)d_in[5];
  const float*     b2  = (const float*)d_in[6];
  const float*     W2r = (const float*)d_in[7];
  const float*     W3l = (const float*)d_in[8];
  const float*     b3  = (const float*)d_in[9];
  const float*     W3r = (const float*)d_in[10];
  const float*     Wfc = (const float*)d_in[11];
  const float*     bfc = (const float*)d_in[12];
  float*           out = (float*)d_out;

  const long long* src = ei;            // edge_index[0]
  const long long* tgt = ei + N_EDGES;  // edge_index[1]

  // Workspace (fp32): hl | hr | agg | hcur (N*16 each) | deg (N) | pbl | pbr
  float* hl   = (float*)d_ws;
  float* hr   = hl + (size_t)N_NODES * HID;
  float* agg  = hr + (size_t)N_NODES * HID;
  float* hcur = agg + (size_t)N_NODES * HID;
  float* deg  = hcur + (size_t)N_NODES * HID;
  float* pbl  = deg + N_NODES;
  float* pbr  = pbl + (size_t)(N_NODES / 4) * 64;  // max K = N_NODES

  const int nFeat   = N_NODES * HID;              // 262144
  const int gFeat   = (nFeat + 255) / 256;
  const int gEdgeF  = (N_EDGES * HID + 255) / 256;
  const int gEdge   = (N_EDGES + 255) / 256;
  const int mTiles  = N_NODES / 16;               // 1024
  const int nPack1  = (N_NODES / 4) * 64;         // K=16384 packed size
  const int nPack2  = (HID / 4) * 64;             // K=16 packed size

  // Degree (shared by all three layers)
  zero_f32<<<(N_NODES + 255) / 256, 256, 0, stream>>>(deg, N_NODES);
  degree_kernel<<<gEdge, 256, 0, stream>>>(tgt, deg, N_EDGES);

  // ---- Layer 1: K = 16384, 8 waves/tile splitting K ----
  pack_b_kernel<<<(nPack1 + 255) / 256, 256, 0, stream>>>(W1l, pbl, N_NODES);
  pack_b_kernel<<<(nPack1 + 255) / 256, 256, 0, stream>>>(W1r, pbr, N_NODES);
  dual_gemm16_wmma<<<mTiles, 256, 0, stream>>>(x, pbl, pbr, hl, hr, N_NODES);
  zero_f32<<<gFeat, 256, 0, stream>>>(agg, nFeat);
  scatter_add_kernel<<<gEdgeF, 256, 0, stream>>>(hl, src, tgt, agg, N_EDGES);
  combine_kernel<<<gFeat, 256, 0, stream>>>(agg, deg, b1, hr, hcur, 1, nFeat);

  // ---- Layer 2: K = 16, single wave per tile ----
  pack_b_kernel<<<1, 256, 0, stream>>>(W2l, pbl, HID);
  pack_b_kernel<<<1, 256, 0, stream>>>(W2r, pbr, HID);
  dual_gemm16_wmma<<<mTiles, 32, 0, stream>>>(hcur, pbl, pbr, hl, hr, HID);
  zero_f32<<<gFeat, 256, 0, stream>>>(agg, nFeat);
  scatter_add_kernel<<<gEdgeF, 256, 0, stream>>>(hl, src, tgt, agg, N_EDGES);
  combine_kernel<<<gFeat, 256, 0, stream>>>(agg, deg, b2, hr, hcur, 1, nFeat);

  // ---- Layer 3: K = 16, no ReLU ----
  pack_b_kernel<<<1, 256, 0, stream>>>(W3l, pbl, HID);
  pack_b_kernel<<<1, 256, 0, stream>>>(W3r, pbr, HID);
  dual_gemm16_wmma<<<mTiles, 32, 0, stream>>>(hcur, pbl, pbr, hl, hr, HID);
  zero_f32<<<gFeat, 256, 0, stream>>>(agg, nFeat);
  scatter_add_kernel<<<gEdgeF, 256, 0, stream>>>(hl, src, tgt, agg, N_EDGES);
  combine_kernel<<<gFeat, 256, 0, stream>>>(agg, deg, b3, hr, hcur, 0, nFeat);

  // ---- Final FC ----
  fc_kernel<<<(N_NODES * R_OUT + 255) / 256, 256, 0, stream>>>(hcur, Wfc, bfc, out);

  (void)in_sizes; (void)n_in; (void)out_size; (void)ws_size;
}